// VNDGCNN_37297495999068
// MI455X (gfx1250) — compile-verified
//
#include <hip/hip_runtime.h>
#include <hip/hip_bf16.h>
#include <math.h>

#define BATCH 16
#define NPTS  2048
#define KNN   20
#define CPB   (NPTS * KNN)          // 40960 columns per batch
#define TILES_PER_WAVE 8
#define LDS_PITCH 132               // 128 rows + 4 pad dwords (bank-conflict break)

typedef float v2f __attribute__((ext_vector_type(2)));
typedef float v4f __attribute__((ext_vector_type(4)));
typedef float v8f __attribute__((ext_vector_type(8)));

// Opaque value laundering: empty asm volatile can't be hoisted/CSE'd, so it
// bounds the scheduler's load-batching window (prevents batch-then-spill).
// Laundering an *offset* (not the pointer) preserves address-space inference,
// so weight fetches stay global_load (LOADcnt) instead of flat_load
// (LOADcnt+DScnt, contends with the LDS datapath).
__device__ __forceinline__ unsigned launder_s(unsigned x) {
    asm volatile("" : "+s"(x));                       // uniform -> SGPR
    return x;
}
__device__ __forceinline__ unsigned launder_v(unsigned x) {
    asm volatile("" : "+v"(x));                       // per-lane -> VGPR
    return x;
}

// Packed-operand channel permutation: channel k lives at
//   pos(k) = (k>>3)*8 + hi_r*4 + p,  hi_r=(k>>1)&1, p=((k>>2)<<1)|(k&1)
__device__ __forceinline__ int pack_pos(int k) {
    const int s2  = k >> 3;
    const int rem = k & 7;
    const int hi_r = (rem >> 1) & 1;
    const int p    = ((rem >> 2) << 1) | (rem & 1);
    return s2 * 8 + hi_r * 4 + p;
}

// ---------------------------------------------------------------------------
// Kernel 1: kNN (top-20 by -squared-distance, self included, index tie-break)
// ---------------------------------------------------------------------------
__global__ __launch_bounds__(256) void knn_kernel(const float* __restrict__ x,
                                                  int* __restrict__ idx) {
    __shared__ float xs[3 * NPTS];
    const int b     = blockIdx.x >> 3;
    const int ntile = blockIdx.x & 7;
    const int tid   = threadIdx.x;

    const float* xb = x + (size_t)b * 3 * NPTS;
    for (int i = tid; i < 3 * NPTS; i += 256) xs[i] = xb[i];
    __syncthreads();

    const int n = ntile * 256 + tid;
    const float px = xs[n], py = xs[NPTS + n], pz = xs[2 * NPTS + n];

    float d[KNN];
    int   id[KNN];
#pragma unroll
    for (int j = 0; j < KNN; ++j) { d[j] = -3.4e38f; id[j] = 0; }

    for (int m = 0; m < NPTS; ++m) {
        const float dx = xs[m] - px;
        const float dy = xs[NPTS + m] - py;
        const float dz = xs[2 * NPTS + m] - pz;
        const float pd = -(fmaf(dx, dx, fmaf(dy, dy, dz * dz)));
        if (pd > d[KNN - 1]) {
            float cd = pd; int ci = m;
#pragma unroll
            for (int j = 0; j < KNN; ++j) {           // sorted-descending insert
                const bool c = cd > d[j];
                const float td = d[j]; const int ti = id[j];
                d[j]  = c ? cd : td;  id[j] = c ? ci : ti;
                cd    = c ? td : cd;  ci    = c ? ti : ci;
            }
        }
    }
    int* op = idx + ((size_t)b * NPTS + n) * KNN;
#pragma unroll
    for (int j = 0; j < KNN; ++j) op[j] = id[j];
}

// ---------------------------------------------------------------------------
// Kernel 2: layer-1 factorization, written in WMMA-packed channel order.
//   P = W1[:, :3] @ x ;  Q = (W1[:,3:] - W1[:, :3]) @ x   (64 ch per point)
// ---------------------------------------------------------------------------
__global__ __launch_bounds__(256) void pq_kernel(const float* __restrict__ x,
                                                 const float* __restrict__ W1,
                                                 float* __restrict__ P,
                                                 float* __restrict__ Q) {
    const int gid = blockIdx.x * 256 + threadIdx.x;
    const int o   = gid & 63;                         // channel
    const int pnt = gid >> 6;
    const int n   = pnt & (NPTS - 1);
    const int b   = pnt >> 11;

    const float* xb = x + (size_t)b * 3 * NPTS;
    const float x0 = xb[n], x1 = xb[NPTS + n], x2 = xb[2 * NPTS + n];

    const float w0 = W1[o * 6 + 0], w1 = W1[o * 6 + 1], w2 = W1[o * 6 + 2];
    const float w3 = W1[o * 6 + 3], w4 = W1[o * 6 + 4], w5 = W1[o * 6 + 5];

    const size_t dst = ((size_t)pnt << 6) + pack_pos(o);
    P[dst] = fmaf(w0, x0, fmaf(w1, x1, w2 * x2));
    Q[dst] = fmaf(w3 - w0, x0, fmaf(w4 - w1, x1, (w5 - w2) * x2));
}

// ---------------------------------------------------------------------------
// Kernel 2b: pre-permute W2 (128x64) / W3 (256x128) into
//   Wp[rt][s2][lane][4];  one b128 load feeds two v_wmma K-steps.
// ---------------------------------------------------------------------------
__global__ __launch_bounds__(256) void wpack_kernel(const float* __restrict__ W,
                                                    float* __restrict__ Wp,
                                                    int kdim) {            // 64 or 128
    const int gid  = blockIdx.x * 256 + threadIdx.x;
    const int e    = gid & 3;
    const int lane = (gid >> 2) & 31;
    const int ns2  = kdim >> 3;
    const int s2   = (gid >> 7) % ns2;
    const int rt   = (gid >> 7) / ns2;
    const int row  = rt * 16 + (lane & 15);
    const int k    = s2 * 8 + (lane >> 4) * 2 + (e & 1) + ((e >> 1) << 2);
    Wp[gid] = W[(size_t)row * kdim + k];
}

// ---------------------------------------------------------------------------
// Kernel 3: fused  h2 = relu(W2 @ h1), h3 = relu(W3 @ h2), max over columns.
// fp32 WMMA 16x16x4, packed operands; laundered per-chunk operand offsets
// bound the load window (no spills, no VGPR-MSB churn, global_load path).
// ---------------------------------------------------------------------------
__global__ __launch_bounds__(128)
__attribute__((amdgpu_waves_per_eu(4)))
void dgcnn_main_kernel(
        const float* __restrict__ W2p, const float* __restrict__ W3p,
        const int* __restrict__ idx, const float* __restrict__ P,
        const float* __restrict__ Q, float* __restrict__ pool) {
    __shared__ float h2s[4 * 16 * LDS_PITCH];         // 33,792 B

    const int tid  = threadIdx.x;
    const int wave = tid >> 5;
    const int lane = tid & 31;
    const int col  = lane & 15;                       // N (B-op) or M (A-op) index
    const int hi   = lane >> 4;                       // K-half selector
    float* myh2 = &h2s[wave * 16 * LDS_PITCH];

    const int w  = blockIdx.x * 4 + wave;             // 0..5119
    const int b  = w / 320;
    const int tb = (w % 320) * TILES_PER_WAVE;

    const v4f* w2q = (const v4f*)W2p;                 // [rt*8  + s2][lane]
    const v4f* w3q = (const v4f*)W3p;                 // [rt*16 + s2][lane]

    v8f maxacc[16];                                   // 256 running row-maxima
#pragma unroll
    for (int rt = 0; rt < 16; ++rt) maxacc[rt] = (v8f){};

    for (int t = 0; t < TILES_PER_WAVE; ++t) {
        const int c = (tb + t) * 16 + col;            // column id in [0, 40960)
        const unsigned n = (unsigned)c / KNN;
        const int m = idx[(size_t)b * CPB + c];
        const v4f* pp = (const v4f*)(P + (((size_t)b * NPTS + m) << 6));
        const v4f* qp = (const v4f*)(Q + (((size_t)b * NPTS + n) << 6));

        // ---- H1 B-tiles (64 x 16): gather + add, packed (16 b128 loads)
        v4f b1q[8];
#pragma unroll
        for (int s2 = 0; s2 < 8; ++s2) {
            const v4f pv = pp[s2 * 2 + hi];
            const v4f qv = qp[s2 * 2 + hi];
            b1q[s2] = pv + qv;
        }

        // ---- layer 2: h2(128x16) = relu(W2 @ H1) -> pre-swizzled LDS
        for (int rt = 0; rt < 8; ++rt) {
            const v4f* wb = w2q + launder_s((unsigned)(rt * 8 * 32));
            v8f acc = (v8f){};
#pragma unroll
            for (int s2 = 0; s2 < 8; ++s2) {
                const v4f av = wb[s2 * 32 + lane];
                acc = __builtin_amdgcn_wmma_f32_16x16x4_f32(
                        false, (v2f){av.x, av.y}, false,
                        (v2f){b1q[s2].x, b1q[s2].y}, (short)0, acc, false, false);
                acc = __builtin_amdgcn_wmma_f32_16x16x4_f32(
                        false, (v2f){av.z, av.w}, false,
                        (v2f){b1q[s2].z, b1q[s2].w}, (short)0, acc, false, false);
            }
#pragma unroll
            for (int j = 0; j < 8; ++j) acc[j] = fmaxf(acc[j], 0.0f);
            // pre-swizzle so the read side can use b128: group (rt*2+hi)
            float* dst = myh2 + col * LDS_PITCH + (rt * 2 + hi) * 8;
            *(v4f*)(dst)     = (v4f){acc[0], acc[1], acc[4], acc[5]};
            *(v4f*)(dst + 4) = (v4f){acc[2], acc[3], acc[6], acc[7]};
        }

        // intra-wave LDS store->load handoff (DS in-order; fence compiler)
        asm volatile("s_wait_dscnt 0x0" ::: "memory");

        // ---- layer 3: h3(256x16) = relu(W3 @ h2); B streams from LDS
#pragma unroll
        for (int rt = 0; rt < 16; ++rt) {
            v8f acc = (v8f){};
#pragma unroll
            for (int h = 0; h < 2; ++h) {             // two laundered K-chunks
                const v4f* wb = w3q + launder_s((unsigned)((rt * 16 + h * 8) * 32));
                const unsigned ro = launder_v(col * LDS_PITCH + hi * 4 + h * 64);
#pragma unroll
                for (int s2 = 0; s2 < 8; ++s2) {
                    const v4f av = wb[s2 * 32 + lane];
                    const v4f bv = *(const v4f*)(myh2 + ro + s2 * 8);
                    acc = __builtin_amdgcn_wmma_f32_16x16x4_f32(
                            false, (v2f){av.x, av.y}, false,
                            (v2f){bv.x, bv.y}, (short)0, acc, false, false);
                    acc = __builtin_amdgcn_wmma_f32_16x16x4_f32(
                            false, (v2f){av.z, av.w}, false,
                            (v2f){bv.z, bv.w}, (short)0, acc, false, false);
                }
            }
#pragma unroll
            for (int j = 0; j < 8; ++j)
                maxacc[rt][j] = fmaxf(maxacc[rt][j], fmaxf(acc[j], 0.0f));
        }
        // keep next iteration's h2 stores ordered after this tile's reads
        asm volatile("s_wait_dscnt 0x0" ::: "memory");
    }

    // ---- reduce the 16 columns held across each 16-lane half
#pragma unroll
    for (int rt = 0; rt < 16; ++rt) {
#pragma unroll
        for (int j = 0; j < 8; ++j) {
            float v = maxacc[rt][j];
            v = fmaxf(v, __shfl_xor(v, 1, 32));
            v = fmaxf(v, __shfl_xor(v, 2, 32));
            v = fmaxf(v, __shfl_xor(v, 4, 32));
            v = fmaxf(v, __shfl_xor(v, 8, 32));
            maxacc[rt][j] = v;
        }
    }
    // all values >= 0 (post-ReLU), so int-bit atomicMax == float max
    if (col == 0) {
#pragma unroll
        for (int rt = 0; rt < 16; ++rt)
#pragma unroll
            for (int j = 0; j < 8; ++j)
                atomicMax((int*)pool + b * 256 + rt * 16 + hi * 8 + j,
                          __float_as_int(maxacc[rt][j]));
    }
}

// ---------------------------------------------------------------------------
// Kernel 4: final FC  out(16x10) = pool(16x256) @ fc_w^T + fc_b
// ---------------------------------------------------------------------------
__global__ void fc_kernel(const float* __restrict__ pool,
                          const float* __restrict__ fcw,
                          const float* __restrict__ fcb,
                          float* __restrict__ out) {
    const int t = threadIdx.x;
    if (t >= BATCH * 10) return;
    const int b = t / 10, cls = t % 10;
    float s = fcb[cls];
    const float* pr = pool + b * 256;
    const float* wr = fcw + cls * 256;
    for (int o = 0; o < 256; ++o) s = fmaf(pr[o], wr[o], s);
    out[t] = s;
}

// ---------------------------------------------------------------------------
extern "C" void kernel_launch(void* const* d_in, const int* in_sizes, int n_in,
                              void* d_out, int out_size, void* d_ws, size_t ws_size,
                              hipStream_t stream) {
    const float* x    = (const float*)d_in[0];
    const float* W1   = (const float*)d_in[1];
    const float* W2   = (const float*)d_in[2];
    const float* W3   = (const float*)d_in[3];
    const float* fc_w = (const float*)d_in[4];
    const float* fc_b = (const float*)d_in[5];

    char* ws = (char*)d_ws;
    const size_t IDX_BYTES = (size_t)BATCH * CPB * sizeof(int);         // 2,621,440
    const size_t PQ_BYTES  = (size_t)BATCH * NPTS * 64 * sizeof(float); // 8,388,608
    int*   idx  = (int*)ws;
    float* P    = (float*)(ws + IDX_BYTES);
    float* Q    = (float*)(ws + IDX_BYTES + PQ_BYTES);
    float* pool = (float*)(ws + IDX_BYTES + 2 * PQ_BYTES);              // 16 KB
    float* W2p  = (float*)(ws + IDX_BYTES + 2 * PQ_BYTES + 16384);      // 32 KB
    float* W3p  = (float*)(ws + IDX_BYTES + 2 * PQ_BYTES + 16384 + 32768); // 128 KB

    hipMemsetAsync(pool, 0, (size_t)BATCH * 256 * sizeof(float), stream);

    knn_kernel<<<BATCH * (NPTS / 256), 256, 0, stream>>>(x, idx);
    pq_kernel<<<(BATCH * NPTS * 64) / 256, 256, 0, stream>>>(x, W1, P, Q);
    wpack_kernel<<<(128 * 64) / 256, 256, 0, stream>>>(W2, W2p, 64);
    wpack_kernel<<<(256 * 128) / 256, 256, 0, stream>>>(W3, W3p, 128);
    // waves = 1280 blocks * 4 = 5120 = 16 batches * 320 units * 8 tiles * 16 cols
    dgcnn_main_kernel<<<1280, 128, 0, stream>>>(W2p, W3p, idx, P, Q, pool);
    fc_kernel<<<1, 256, 0, stream>>>(pool, fc_w, fc_b, (float*)d_out);
}